// _RoIOffsetPooling_86320252715839
// MI455X (gfx1250) — compile-verified
//
#include <hip/hip_runtime.h>

typedef __attribute__((ext_vector_type(2))) float v2f;
typedef __attribute__((ext_vector_type(8))) float v8f;

#define PH_ 7
#define PW_ 7
#define SS_ 0.0625f
#define TRANS_STD_ 0.1f
#define CC_ 256
#define HH_ 168
#define WW_ 168
#define NROI_ 512

__global__ __launch_bounds__(128)
void deform_roi_pool_wmma(const float* __restrict__ feat,
                          const float* __restrict__ rois,
                          const float* __restrict__ offset,
                          float* __restrict__ out) {
  const int bin = blockIdx.x;                 // 0 .. NROI*49-1
  const int n   = bin / (PH_ * PW_);
  const int phw = bin % (PH_ * PW_);
  const int ph  = phw / PW_;
  const int pw  = phw % PW_;
  const int wave = threadIdx.x >> 5;          // 0..3  (64 channels each)
  const int lane = threadIdx.x & 31;
  const bool hi  = (lane >= 16);              // K-half: corners {2,3} vs {0,1}

  // ---- RoI geometry (uniform across the block) ----
  const float* r = rois + n * 5;
  const int   b  = (int)r[0];
  const float x1 = r[1], y1 = r[2], x2 = r[3], y2 = r[4];
  const float sw = x1 * SS_ - 0.5f;
  const float sh = y1 * SS_ - 0.5f;
  const float rw = fmaxf((x2 + 1.0f) * SS_ - 0.5f - sw, 0.1f);
  const float rh = fmaxf((y2 + 1.0f) * SS_ - 0.5f - sh, 0.1f);
  const float bin_w = rw * (1.0f / PW_);
  const float bin_h = rh * (1.0f / PH_);
  const float sub_w = bin_w * 0.5f;
  const float sub_h = bin_h * 0.5f;
  const float offx = offset[((n * 2 + 0) * PH_ + ph) * PW_ + pw] * TRANS_STD_;
  const float offy = offset[((n * 2 + 1) * PH_ + ph) * PW_ + pw] * TRANS_STD_;

  // ---- Per-lane (weight, offset) pairs. WMMA #s contracts the 4 corners of
  // subsample s: lanes 0-15 carry the y0 row (corners 0,1 -> local K 0,1),
  // lanes 16-31 carry the y1i row (corners 2,3 -> local K 2,3). ----
  float w0s[4], w1s[4];
  int   o0s[4], o1s[4];
  float cnt = 0.0f;
#pragma unroll
  for (int s = 0; s < 4; ++s) {
    const int ihh = s >> 1, iww = s & 1;
    const float x = sw + (float)pw * bin_w + offx * rw + ((float)iww + 0.5f) * sub_w;
    const float y = sh + (float)ph * bin_h + offy * rh + ((float)ihh + 0.5f) * sub_h;
    const bool valid = (x > -0.5f) && (x < (float)WW_ - 0.5f) &&
                       (y > -0.5f) && (y < (float)HH_ - 0.5f);
    const float xc = fminf(fmaxf(x, 0.0f), (float)(WW_ - 1));
    const float yc = fminf(fmaxf(y, 0.0f), (float)(HH_ - 1));
    const float x0f = floorf(xc), y0f = floorf(yc);
    const float lx = xc - x0f, ly = yc - y0f;
    const int x0  = (int)x0f, y0 = (int)y0f;
    const int x1i = min(x0 + 1, WW_ - 1);
    const int y1i = min(y0 + 1, HH_ - 1);
    const float v = valid ? 1.0f : 0.0f;
    cnt += v;
    const float wy  = (hi ? ly : (1.0f - ly)) * v;   // validity folded in
    const int   row = hi ? y1i : y0;
    o0s[s] = row * WW_ + x0;
    o1s[s] = row * WW_ + x1i;
    w0s[s] = wy * (1.0f - lx);
    w1s[s] = wy * lx;
  }
  // Fold 1/count into the weights (cnt==0 -> all weights 0 -> output 0, as ref).
  const float inv = (cnt > 0.0f) ? (1.0f / cnt) : 0.0f;
#pragma unroll
  for (int s = 0; s < 4; ++s) { w0s[s] *= inv; w1s[s] *= inv; }

  // ---- WMMA contraction: out[c] = sum_k wgt[k] * feat[b, c, ofs[k]] ----
  const float* fb = feat + (size_t)b * ((size_t)CC_ * HH_ * WW_);
  const int cbase = wave * 64;

#pragma unroll
  for (int ch = 0; ch < 4; ++ch) {
    const int c = cbase + ch * 16 + (lane & 15);    // A-matrix row M
    const float* fc = fb + (size_t)c * (HH_ * WW_);
    v8f acc = {};
#pragma unroll
    for (int ks = 0; ks < 4; ++ks) {
      v2f a, bw;
      a.x  = fc[o0s[ks]];  a.y  = fc[o1s[ks]];
      bw.x = w0s[ks];      bw.y = w1s[ks];
      acc = __builtin_amdgcn_wmma_f32_16x16x4_f32(
          /*neg_a=*/false, a, /*neg_b=*/false, bw,
          /*c_mod=*/(short)0, acc, /*reuse_a=*/false, /*reuse_b=*/false);
    }
    // D layout: VGPR i -> M=i (lanes 0-15) / M=8+i (lanes 16-31), N-replicated.
    // Lanes 0-7 store channels cbase+ch*16+0..7; lanes 16-23 store +8..15.
    const int m = lane & 7;
    float v = acc[0];
    if (m == 1) v = acc[1];
    if (m == 2) v = acc[2];
    if (m == 3) v = acc[3];
    if (m == 4) v = acc[4];
    if (m == 5) v = acc[5];
    if (m == 6) v = acc[6];
    if (m == 7) v = acc[7];
    if ((lane & 15) < 8) {
      const int cc = cbase + ch * 16 + (hi ? 8 : 0) + m;
      out[((size_t)n * CC_ + cc) * (PH_ * PW_) + phw] = v;
    }
  }
}

extern "C" void kernel_launch(void* const* d_in, const int* in_sizes, int n_in,
                              void* d_out, int out_size, void* d_ws, size_t ws_size,
                              hipStream_t stream) {
  (void)in_sizes; (void)n_in; (void)d_ws; (void)ws_size; (void)out_size;
  const float* feat   = (const float*)d_in[0];  // (8,256,168,168)
  const float* rois   = (const float*)d_in[1];  // (512,5)
  const float* offset = (const float*)d_in[2];  // (512,2,7,7)
  float* out = (float*)d_out;                   // (512,256,7,7)

  const int bins = NROI_ * PH_ * PW_;           // 25088
  deform_roi_pool_wmma<<<dim3(bins), dim3(128), 0, stream>>>(feat, rois, offset, out);
}